// CrossEntropyLoss_3624952398573
// MI455X (gfx1250) — compile-verified
//
#include <hip/hip_runtime.h>
#include <stdint.h>

// -------- problem constants (match reference) --------
#define BATCH   4
#define NSRC    4096          // S_SRC * L_SRC
#define NTGT    4096          // S_TGT * L_TGT
#define STGT    8
#define LTGT    512
#define PADBITS 0xBF800000u   // bit pattern of -1.0f
#define INV_TAU 10.0f
#define EPS_P   1e-15f

#define CHUNK_ROWS   32       // rows per block
#define CHUNKS_PER_B (NTGT / CHUNK_ROWS)   // 128
#define GRID_MAIN    (BATCH * CHUNKS_PER_B) // 512

typedef unsigned int v4u __attribute__((ext_vector_type(4)));
typedef float        v4f __attribute__((ext_vector_type(4)));
typedef int          v8i __attribute__((ext_vector_type(8)));
typedef int          v4i __attribute__((ext_vector_type(4)));

__device__ __forceinline__ float wave_max(float v) {
    #pragma unroll
    for (int o = 16; o > 0; o >>= 1) v = fmaxf(v, __shfl_xor(v, o, 32));
    return v;
}
__device__ __forceinline__ float wave_sum(float v) {
    #pragma unroll
    for (int o = 16; o > 0; o >>= 1) v += __shfl_xor(v, o, 32);
    return v;
}
__device__ __forceinline__ int wave_or(int v) {
    #pragma unroll
    for (int o = 16; o > 0; o >>= 1) v |= __shfl_xor(v, o, 32);
    return v;
}

// Exact color equality: compare (x,y) key bits; on the rare hit (~1/97),
// confirm (z,w) from the hot 64 KB src-color table in cache.
__device__ __forceinline__ bool eq_full(unsigned kx, unsigned ky, const uint4 tk,
                                        const uint2* __restrict__ srcK, int j) {
    bool eq = (kx == tk.x) & (ky == tk.y);
    if (eq) {
        const uint2 zw = srcK[2 * j + 1];
        eq = (zw.x == tk.z) & (zw.y == tk.w);
    }
    return eq;
}

// Per-row kernel: masked softmax over src columns, gt-color mass, nll.
__global__ __launch_bounds__(256) void seg_ce_rows(
    const float* __restrict__ sim,      // [B, NTGT, NSRC]
    const float* __restrict__ srcc,     // [B, NSRC, 4]
    const float* __restrict__ tgtc,     // [B, NTGT, 4]
    float* __restrict__ wnll,           // [B*NTGT] (masked nll)
    float* __restrict__ wm)             // [B*NTGT] (matched flag)
{
    __shared__ __align__(16) uint2 sXY[NSRC];   // 32 KB: (x,y) bit-keys of src colors

    const int b     = blockIdx.x >> 7;          // / CHUNKS_PER_B
    const int chunk = blockIdx.x & (CHUNKS_PER_B - 1);
    const int lane  = threadIdx.x & 31;
    const int wid   = threadIdx.x >> 5;

    // src colors of batch b, viewed as uint2 pairs: color j = {xy at [2j], zw at [2j+1]}
    const uint2* srcK = reinterpret_cast<const uint2*>(srcc) + (size_t)b * NSRC * 2;

    // ---- Stage (x,y) keys into LDS via the Tensor Data Mover -------------
    // Strided compaction: 4096 tiles of one 8-byte element, memory stride 16 B.
#if __has_builtin(__builtin_amdgcn_tensor_load_to_lds)
    if (wid == 0) {
        const unsigned long long ga = (unsigned long long)(uintptr_t)srcK;
        const unsigned lds = (unsigned)(uintptr_t)(&sXY[0]); // low 32 bits = LDS offset
        v4u g0 = { 1u,                                   // count=1 valid descriptor
                   lds,                                  // lds_addr
                   (unsigned)(ga & 0xFFFFFFFFu),
                   (unsigned)((ga >> 32) & 0x1FFFFFFu) | (2u << 30) }; // addr[56:32] | type=2
        v8i g1 = { 0x00030000,   // data_size=3 (8B), wg_mask=0, no pad/iterate
                   0x00020000,   // tensor_dim0 = 2 elements (lo16<<16), atomic_barrier_addr=0
                   0x10000000,   // tensor_dim0 hi=0 | tensor_dim1 lo16 = 4096
                   0x00010000,   // tensor_dim1 hi=0 | tile_dim0 = 1
                   0x00001000,   // tile_dim1 = 4096 | tile_dim2 = 0
                   2,            // tensor_dim0_stride lo32 = 2 elements (16 B)
                   0x00020000,   // stride0 hi=0 | tensor_dim1_stride lo16 = 2
                   0 };          // tensor_dim1_stride hi = 0
        v4i z4 = { 0, 0, 0, 0 };
        v8i z8 = { 0, 0, 0, 0, 0, 0, 0, 0 };
        __builtin_amdgcn_tensor_load_to_lds(g0, g1, z4, z4, z8, 0);
        __builtin_amdgcn_s_wait_tensorcnt(0);
    }
    __syncthreads();
#else
    for (int j = threadIdx.x; j < NSRC; j += 256) sXY[j] = srcK[2 * j];
    __syncthreads();
#endif

    const uint4* tgtK = reinterpret_cast<const uint4*>(tgtc);

    #pragma unroll 1
    for (int r = 0; r < CHUNK_ROWS / 8; ++r) {       // 4 rows per wave
        const int    t    = chunk * CHUNK_ROWS + r * 8 + wid;
        const size_t rowi = (size_t)b * NTGT + t;
        const float* row  = sim + (rowi << 12);      // * NSRC
        const uint4  tk   = tgtK[rowi];
        const bool   vt   = (tk.x != PADBITS);       // valid palette colors are in [0,1)

        // ---- pass A: row max over valid columns + matched detection ----
        // 512 B contiguous per wave VMEM instruction (b128 per lane).
        float mx  = -3.0e38f;
        int   any = 0;
        #pragma unroll 2
        for (int j4 = lane * 4; j4 < NSRC; j4 += 128) {
            const v4f x = *(const v4f*)(row + j4);
            const v4u a = *(const v4u*)(sXY + j4);       // keys j4+0, j4+1
            const v4u c = *(const v4u*)(sXY + j4 + 2);   // keys j4+2, j4+3
            if (a.x != PADBITS) mx = fmaxf(mx, x.x);
            if (a.z != PADBITS) mx = fmaxf(mx, x.y);
            if (c.x != PADBITS) mx = fmaxf(mx, x.z);
            if (c.z != PADBITS) mx = fmaxf(mx, x.w);
            any |= (int)eq_full(a.x, a.y, tk, srcK, j4 + 0);
            any |= (int)eq_full(a.z, a.w, tk, srcK, j4 + 1);
            any |= (int)eq_full(c.x, c.y, tk, srcK, j4 + 2);
            any |= (int)eq_full(c.z, c.w, tk, srcK, j4 + 3);
        }
        mx  = wave_max(mx);
        any = wave_or(any) & (int)vt;                // matched = valid_tgt && any-eq

        if (!any) {                                  // row excluded by nll mask
            if (lane == 0) { wnll[rowi] = 0.0f; wm[rowi] = 0.0f; }
            continue;
        }

        // ---- pass B: softmax sums (row hot in L2; last-use NT loads) ----
        const float M  = mx * INV_TAU;
        float sa = 0.0f, se = 0.0f;
        #pragma unroll 2
        for (int j4 = lane * 4; j4 < NSRC; j4 += 128) {
            const v4f x = __builtin_nontemporal_load((const v4f*)(row + j4));
            const v4u a = *(const v4u*)(sXY + j4);
            const v4u c = *(const v4u*)(sXY + j4 + 2);
            #pragma unroll
            for (int i = 0; i < 4; ++i) {
                const unsigned kx = (i == 0) ? a.x : (i == 1) ? a.z : (i == 2) ? c.x : c.z;
                const unsigned ky = (i == 0) ? a.y : (i == 1) ? a.w : (i == 2) ? c.y : c.w;
                if (kx != PADBITS) {
                    const float e = __expf(__fmaf_rn(x[i], INV_TAU, -M));
                    sa += e;
                    se += eq_full(kx, ky, tk, srcK, j4 + i) ? e : 0.0f;
                }
            }
        }
        sa = wave_sum(sa);
        se = wave_sum(se);

        if (lane == 0) {
            const float p = se / sa;
            wnll[rowi] = -__logf(p + EPS_P);
            wm[rowi]   = 1.0f;
        }
    }
}

// Finisher: segment means -> total loss + segment count. 16 waves, 2 segs/wave.
__global__ __launch_bounds__(512) void seg_ce_finish(
    const float* __restrict__ wnll, const float* __restrict__ wm,
    float* __restrict__ out)
{
    __shared__ float segLoss[BATCH * STGT];
    __shared__ int   segValid[BATCH * STGT];

    const int lane = threadIdx.x & 31;
    const int w    = threadIdx.x >> 5;       // 0..15

    for (int s = w; s < BATCH * STGT; s += 16) {
        float sn = 0.0f, sm = 0.0f;
        for (int i = lane; i < LTGT; i += 32) {
            const int idx = s * LTGT + i;
            sn += wnll[idx];                 // already masked
            sm += wm[idx];
        }
        sn = wave_sum(sn);
        sm = wave_sum(sm);
        if (lane == 0) {
            segValid[s] = (sm > 0.0f) ? 1 : 0;
            segLoss[s]  = (sm > 0.0f) ? (sn / fmaxf(sm, 1.0f)) : 0.0f;
        }
    }
    __syncthreads();

    if (threadIdx.x == 0) {
        float total = 0.0f; int cnt = 0;
        #pragma unroll
        for (int s = 0; s < BATCH * STGT; ++s) { total += segLoss[s]; cnt += segValid[s]; }
        out[0] = total / (float)max(cnt, 1);
        out[1] = (float)cnt;
    }
}

extern "C" void kernel_launch(void* const* d_in, const int* in_sizes, int n_in,
                              void* d_out, int out_size, void* d_ws, size_t ws_size,
                              hipStream_t stream) {
    const float* sim  = (const float*)d_in[0];   // seg_sim_map  [4,4096,4096]
    const float* srcc = (const float*)d_in[1];   // seg_colors_src [4,8,512,4]
    const float* tgtc = (const float*)d_in[2];   // seg_colors_tgt [4,8,512,4]

    float* wnll = (float*)d_ws;                  // 16384 floats
    float* wm   = wnll + BATCH * NTGT;           // 16384 floats

    seg_ce_rows<<<GRID_MAIN, 256, 0, stream>>>(sim, srcc, tgtc, wnll, wm);
    seg_ce_finish<<<1, 512, 0, stream>>>(wnll, wm, (float*)d_out);
}